// Branching_Kriging_73615739453589
// MI455X (gfx1250) — compile-verified
//
#include <hip/hip_runtime.h>
#include <math.h>

// ---------------------------------------------------------------------------
// Branching-Kriging as a WMMA GEMM on gfx1250 (CDNA5, wave32).
//
// K-dim feature layout (KF = 80):
//   [0..7]    share cross terms        A: 2*a_s*X1_s          B: X2_s
//   [8..16]   nested linear cross      A: 2*V1_bm             B: V2_bm
//   [17..76]  per (b in 0..2, z in 1..4), 5 dims each:
//      +0     A: e1*(t_b - sum_m h*V1^2)   B: e2
//      +1..3  A: e1*2*h*V1_bm              B: e2*V2_bm
//      +4     A: e1                        B: -e2*sum_m h*V2^2
//   [77]      A: u_i (row norms + const)   B: 1
//   [78]      A: 1                         B: v_j (col norms)
//   [79]      pad = 0
// Then out[i][j] = exp( dot(A_i, B_j) ).
//
// Main kernel: one wave per 32x32 macro-tile (2x2 WMMA tiles) -> 4 independent
// accumulator chains, 1:1 load:WMMA ratio, 5x read amplification (vs 10x).
// ---------------------------------------------------------------------------

#define S_FACT 8
#define B_FACT 3
#define NB     3          // NEST[b] == 3 for all b
#define NLEV   4          // LEVELS == 4
#define WCOLS  20         // S_FACT + B_FACT + sum(NEST)
#define KF     80

typedef __attribute__((ext_vector_type(2))) float v2f;
typedef __attribute__((ext_vector_type(8))) float v8f;

__device__ __forceinline__ float act_fun(float x) {
    float y = (x >= 0.0f) ? (x + 1.0f) : __expf(x);
    return fminf(y, 30.0f);
}

// One thread per row: build the 80-wide feature vector.
__global__ __launch_bounds__(256)
void bk_build_features(const float* __restrict__ W, int n, int isA,
                       const float* __restrict__ alpha,
                       const float* __restrict__ theta,
                       const float* __restrict__ g0,
                       const float* __restrict__ g1,
                       const float* __restrict__ g2,
                       float* __restrict__ feat) {
    int row = blockIdx.x * blockDim.x + threadIdx.x;
    if (row >= n) return;

    const float* w = W + (long)row * WCOLS;
    float X[S_FACT], Z[B_FACT], V[NB * B_FACT];
#pragma unroll
    for (int s = 0; s < S_FACT; ++s) X[s] = w[s];
#pragma unroll
    for (int b = 0; b < B_FACT; ++b) Z[b] = w[S_FACT + b];
#pragma unroll
    for (int m = 0; m < NB * B_FACT; ++m) V[m] = w[S_FACT + B_FACT + m];

    float a[S_FACT], t[B_FACT];
#pragma unroll
    for (int s = 0; s < S_FACT; ++s) a[s] = act_fun(alpha[s]);
#pragma unroll
    for (int b = 0; b < B_FACT; ++b) t[b] = act_fun(theta[b]);

    float* f = feat + (long)row * KF;

    // Separable norm terms.
    float normX = 0.0f, normV = 0.0f, sumT = 0.0f;
#pragma unroll
    for (int s = 0; s < S_FACT; ++s) normX += a[s] * X[s] * X[s];
#pragma unroll
    for (int m = 0; m < NB * B_FACT; ++m) normV += V[m] * V[m];
#pragma unroll
    for (int b = 0; b < B_FACT; ++b) sumT += t[b];

    if (isA) {
#pragma unroll
        for (int s = 0; s < S_FACT; ++s) f[s] = 2.0f * a[s] * X[s];
#pragma unroll
        for (int m = 0; m < NB * B_FACT; ++m) f[8 + m] = 2.0f * V[m];
    } else {
#pragma unroll
        for (int s = 0; s < S_FACT; ++s) f[s] = X[s];
#pragma unroll
        for (int m = 0; m < NB * B_FACT; ++m) f[8 + m] = V[m];
    }

#pragma unroll
    for (int b = 0; b < B_FACT; ++b) {
        const float* gp = (b == 0) ? g0 : ((b == 1) ? g1 : g2);
#pragma unroll
        for (int zp = 0; zp < NLEV; ++zp) {           // level z = zp+1
            float e = (Z[b] == (float)(zp + 1)) ? 1.0f : 0.0f;
            float h[NB];
            float sumhVV = 0.0f;
#pragma unroll
            for (int m = 0; m < NB; ++m) {
                h[m] = act_fun(gp[m * NLEV + zp]) - 1.0f;   // gact(m, z) - 1
                float vv = V[b * NB + m];
                sumhVV += h[m] * vv * vv;
            }
            int idx0 = 17 + (b * NLEV + zp) * 5;
            if (isA) {
                f[idx0] = e * (t[b] - sumhVV);
#pragma unroll
                for (int m = 0; m < NB; ++m)
                    f[idx0 + 1 + m] = e * 2.0f * h[m] * V[b * NB + m];
                f[idx0 + 4] = e;
            } else {
                f[idx0] = e;
#pragma unroll
                for (int m = 0; m < NB; ++m)
                    f[idx0 + 1 + m] = e * V[b * NB + m];
                f[idx0 + 4] = -e * sumhVV;
            }
        }
    }

    if (isA) {
        f[77] = -normX - normV - sumT;   // u_i (carries the -sum(t) constant)
        f[78] = 1.0f;
    } else {
        f[77] = 1.0f;
        f[78] = -normX - normV;          // v_j
    }
    f[79] = 0.0f;
}

// One wave per 32x32 macro-tile (2x2 WMMA tiles); 8 waves (256 thr) per block.
// 80 x V_WMMA_F32_16X16X4_F32 over KF=80 into 4 independent accumulators,
// then exp + store.
__global__ __launch_bounds__(256)
void bk_wmma_tile(const float* __restrict__ Af,
                  const float* __restrict__ Bf,
                  float* __restrict__ out,
                  int n1, int n2, int nMacro) {
    const int wave   = threadIdx.x >> 5;
    const int lane   = threadIdx.x & 31;
    const int laneLo = lane & 15;
    const int laneHi = lane >> 4;

    int mt = blockIdx.x * 8 + wave;            // wave-uniform macro-tile id
    if (mt >= nMacro) return;                  // whole wave exits: EXEC all-1s

    const int macroPerRow = n2 >> 5;
    const int ti = mt / macroPerRow;           // macro row (32 rows of out)
    const int tj = mt % macroPerRow;           // macro col (32 cols of out)

    // A fragment: lane L, vgpr g -> A[M = L%16][K = kk + 2*(L/16) + g]
    // B fragment: lane L, vgpr g -> B[K = kk + 2*(L/16) + g][N = L%16] = Bf[col][K]
    const float* arow0 = Af + (long)(ti * 32 + laneLo) * KF + 2 * laneHi;
    const float* arow1 = arow0 + 16 * KF;
    const float* brow0 = Bf + (long)(tj * 32 + laneLo) * KF + 2 * laneHi;
    const float* brow1 = brow0 + 16 * KF;

    v8f acc00 = {}, acc01 = {}, acc10 = {}, acc11 = {};
#pragma unroll
    for (int kk = 0; kk < KF; kk += 4) {
        v2f a0 = *(const v2f*)(arow0 + kk);
        v2f a1 = *(const v2f*)(arow1 + kk);
        v2f b0 = *(const v2f*)(brow0 + kk);
        v2f b1 = *(const v2f*)(brow1 + kk);
        acc00 = __builtin_amdgcn_wmma_f32_16x16x4_f32(false, a0, false, b0,
                                                      (short)0, acc00, false, false);
        acc01 = __builtin_amdgcn_wmma_f32_16x16x4_f32(false, a0, false, b1,
                                                      (short)0, acc01, false, false);
        acc10 = __builtin_amdgcn_wmma_f32_16x16x4_f32(false, a1, false, b0,
                                                      (short)0, acc10, false, false);
        acc11 = __builtin_amdgcn_wmma_f32_16x16x4_f32(false, a1, false, b1,
                                                      (short)0, acc11, false, false);
    }

    // C/D layout: vgpr g, lane L -> D[M = g + 8*(L/16)][N = L%16]
    const int col0 = tj * 32 + laneLo;
    const int row0 = ti * 32 + 8 * laneHi;
    float* o00 = out + (long)row0 * n2 + col0;          // tile (0,0)
#pragma unroll
    for (int g = 0; g < 8; ++g) {
        long r = (long)g * n2;
        o00[r]            = __expf(acc00[g]);           // rows row0+g,    cols col0
        o00[r + 16]       = __expf(acc01[g]);           // rows row0+g,    cols col0+16
        o00[r + 16L * n2] = __expf(acc10[g]);           // rows row0+16+g, cols col0
        o00[r + 16L * n2 + 16] = __expf(acc11[g]);      // rows row0+16+g, cols col0+16
    }
}

extern "C" void kernel_launch(void* const* d_in, const int* in_sizes, int n_in,
                              void* d_out, int out_size, void* d_ws, size_t ws_size,
                              hipStream_t stream) {
    const float* W1    = (const float*)d_in[0];
    const float* W2    = (const float*)d_in[1];
    const float* alpha = (const float*)d_in[2];
    const float* theta = (const float*)d_in[3];
    const float* g0    = (const float*)d_in[4];
    const float* g1    = (const float*)d_in[5];
    const float* g2    = (const float*)d_in[6];
    float* out = (float*)d_out;

    const int n1 = in_sizes[0] / WCOLS;   // 4096
    const int n2 = in_sizes[1] / WCOLS;   // 2048

    float* Af = (float*)d_ws;
    float* Bf = Af + (size_t)n1 * KF;

    bk_build_features<<<(n1 + 255) / 256, 256, 0, stream>>>(
        W1, n1, 1, alpha, theta, g0, g1, g2, Af);
    bk_build_features<<<(n2 + 255) / 256, 256, 0, stream>>>(
        W2, n2, 0, alpha, theta, g0, g1, g2, Bf);

    const int nMacro = (n1 >> 5) * (n2 >> 5);       // 32x32 macro-tiles
    const int nBlocks = (nMacro + 7) / 8;
    bk_wmma_tile<<<nBlocks, 256, 0, stream>>>(Af, Bf, out, n1, n2, nMacro);
}